// LGA_49331994362180
// MI455X (gfx1250) — compile-verified
//
#include <hip/hip_runtime.h>
#include <hip/hip_bf16.h>
#include <math.h>

// MI455X / gfx1250, wave32.
typedef float v2f __attribute__((ext_vector_type(2)));
typedef float v8f __attribute__((ext_vector_type(8)));
typedef float f4  __attribute__((ext_vector_type(4)));

// ---------------------------------------------------------------------------
// Streaming copy of knn_x -> out (128 MB). Regular-temporal loads (warm L2 for
// the aggregation kernel that follows), non-temporal stores (output is never
// re-read). 128-bit per lane.
// ---------------------------------------------------------------------------
__global__ void lga_copy_kernel(const f4* __restrict__ src,
                                f4* __restrict__ dst, size_t n4) {
    size_t i = (size_t)blockIdx.x * blockDim.x + threadIdx.x;
    size_t stride = (size_t)gridDim.x * blockDim.x;
    for (; i < n4; i += stride) {
        f4 v = src[i];
        __builtin_nontemporal_store(v, &dst[i]);
    }
}

// ---------------------------------------------------------------------------
// One wave (32 lanes) per point. lane == neighbor k.
//  - bin each neighbor by argmax over the 6 sphere directions
//  - sum_features[6x64] = onehot^T(16x32-pad) x knn_x(32x64) via
//    V_WMMA_F32_16X16X4_F32: 8 K-chunks of 4 neighbors, 4 N-tiles of 16 dims
//  - avg_direction via LDS ds_add_f32 atomics
//  - counts / direction_percentage via wave ballots
// ---------------------------------------------------------------------------
__global__ __launch_bounds__(32)
void lga_point_kernel(const float* __restrict__ lc_xyz,
                      const float* __restrict__ knn_xyz,
                      const float* __restrict__ knn_x,
                      const float* __restrict__ sphere,
                      float* __restrict__ dir_pct,    // [BN,6]
                      float* __restrict__ avg_dir,    // [BN,6,3]
                      float* __restrict__ avg_feat,   // [BN,6,64]
                      float* __restrict__ k_infl) {   // [BN]
    const int pt   = blockIdx.x;
    const int lane = threadIdx.x;              // 0..31

    __shared__ int   s_idx[32];
    __shared__ float s_dir[18];
    __shared__ float s_cnt[6];

    // ---- geometry: rel, dist, bin index (argmax keeps first max, like jnp)
    const float cx = lc_xyz[pt * 3 + 0];
    const float cy = lc_xyz[pt * 3 + 1];
    const float cz = lc_xyz[pt * 3 + 2];
    const float* kp = knn_xyz + ((size_t)pt * 32 + lane) * 3;
    const float rx = kp[0] - cx, ry = kp[1] - cy, rz = kp[2] - cz;
    const float dist = sqrtf(rx * rx + ry * ry + rz * rz);
    const float inv  = 1.0f / (dist + 1e-8f);
    const float nx = rx * inv, ny = ry * inv, nz = rz * inv;

    int best = 0;
    float bestv = -3.402823e38f;
    #pragma unroll
    for (int a = 0; a < 6; ++a) {
        float s = nx * sphere[a * 3 + 0] + ny * sphere[a * 3 + 1] + nz * sphere[a * 3 + 2];
        if (s > bestv) { bestv = s; best = a; }
    }
    s_idx[lane] = best;
    if (lane < 18) s_dir[lane] = 0.0f;
    __syncthreads();   // single-wave WG: lowers to S_NOP + waits, cheap

    // ---- WMMA accumulation: sum_features (bins padded 6->16)
    // A 16x4 f32 layout: lane holds row M=lane&15; VGPR0 = K {0|2}, VGPR1 = K {1|3}
    // B 4x16  f32 layout: lane holds col N=lane&15; same K striping.
    const float* X = knn_x + (size_t)pt * 32 * 64;
    const int m  = lane & 15;
    const int kk = (lane >> 4) << 1;           // 0 for lanes 0-15, 2 for 16-31

    v8f acc[4];
    #pragma unroll
    for (int t = 0; t < 4; ++t) acc[t] = {};

    #pragma unroll
    for (int c = 0; c < 8; ++c) {
        const int k0 = 4 * c + kk;
        v2f a;
        a.x = (s_idx[k0]     == m) ? 1.0f : 0.0f;
        a.y = (s_idx[k0 + 1] == m) ? 1.0f : 0.0f;
        #pragma unroll
        for (int t = 0; t < 4; ++t) {
            const int d = 16 * t + m;          // feature column for this lane
            v2f bm;
            bm.x = X[(size_t)k0 * 64 + d];
            bm.y = X[(size_t)(k0 + 1) * 64 + d];
            acc[t] = __builtin_amdgcn_wmma_f32_16x16x4_f32(
                false, a, false, bm, (short)0, acc[t], false, false);
        }
    }

    // ---- per-bin feature sum over D (bin r lives in C/D VGPR r, lanes 0-15)
    float rowsum[6];
    #pragma unroll
    for (int r = 0; r < 6; ++r) {
        float s = acc[0][r] + acc[1][r] + acc[2][r] + acc[3][r];
        #pragma unroll
        for (int off = 8; off >= 1; off >>= 1)
            s += __shfl_xor(s, off, 32);
        rowsum[r] = s;
    }

    // ---- counts via wave ballots (uniform control flow)
    float c0 = (float)__popcll(__ballot(best == 0)) - 1.0f;  // bin0 excludes center
    float c1 = (float)__popcll(__ballot(best == 1));
    float c2 = (float)__popcll(__ballot(best == 2));
    float c3 = (float)__popcll(__ballot(best == 3));
    float c4 = (float)__popcll(__ballot(best == 4));
    float c5 = (float)__popcll(__ballot(best == 5));

    // ---- scatter-add of rel into per-bin direction sums (ds_add_f32)
    atomicAdd(&s_dir[best * 3 + 0], rx);
    atomicAdd(&s_dir[best * 3 + 1], ry);
    atomicAdd(&s_dir[best * 3 + 2], rz);
    if (lane == 0) {
        s_cnt[0] = c0; s_cnt[1] = c1; s_cnt[2] = c2;
        s_cnt[3] = c3; s_cnt[4] = c4; s_cnt[5] = c5;
    }
    __syncthreads();

    // ---- avg_features: acc[t][r] / (rowsum[r] + 1e-9)
    if (lane < 16) {
        #pragma unroll
        for (int r = 0; r < 6; ++r) {
            const float invs = 1.0f / (rowsum[r] + 1e-9f);
            #pragma unroll
            for (int t = 0; t < 4; ++t)
                avg_feat[((size_t)pt * 6 + r) * 64 + 16 * t + lane] = acc[t][r] * invs;
        }
    }

    // ---- avg_direction = sum_direction / (counts + 1e-8)
    if (lane < 18) {
        const int mb = lane / 3;
        avg_dir[(size_t)pt * 18 + lane] = s_dir[lane] / (s_cnt[mb] + 1e-8f);
    }

    // ---- direction_percentage: threshold counts > ALPHA(=2), normalize
    const float t0 = c0 > 2.0f ? c0 : 0.0f;
    const float t1 = c1 > 2.0f ? c1 : 0.0f;
    const float t2 = c2 > 2.0f ? c2 : 0.0f;
    const float t3 = c3 > 2.0f ? c3 : 0.0f;
    const float t4 = c4 > 2.0f ? c4 : 0.0f;
    const float t5 = c5 > 2.0f ? c5 : 0.0f;
    const float invt = 1.0f / (t0 + t1 + t2 + t3 + t4 + t5 + 1e-8f);
    if (lane == 0) {
        float* dp = dir_pct + (size_t)pt * 6;
        dp[0] = t0 * invt; dp[1] = t1 * invt; dp[2] = t2 * invt;
        dp[3] = t3 * invt; dp[4] = t4 * invt; dp[5] = t5 * invt;
        k_infl[pt] = 1.0f;
    }
}

extern "C" void kernel_launch(void* const* d_in, const int* in_sizes, int n_in,
                              void* d_out, int out_size, void* d_ws, size_t ws_size,
                              hipStream_t stream) {
    (void)n_in; (void)out_size; (void)d_ws; (void)ws_size;
    const float* lc_xyz  = (const float*)d_in[0];
    const float* lc_x    = (const float*)d_in[1];  // unused by the reference
    const float* knn_xyz = (const float*)d_in[2];
    const float* knn_x   = (const float*)d_in[3];
    const float* sphere  = (const float*)d_in[4];
    (void)lc_x;

    const size_t BN      = (size_t)in_sizes[0] / 3;    // B*N = 16384
    const size_t szKnnX  = (size_t)in_sizes[3];        // B*N*32*64

    float* out       = (float*)d_out;
    float* out_knnx  = out;                             // [BN,32,64]
    float* out_dp    = out_knnx + szKnnX;               // [BN,6]
    float* out_ad    = out_dp + BN * 6;                 // [BN,6,3]
    float* out_af    = out_ad + BN * 6 * 3;             // [BN,6,64]
    float* out_ki    = out_af + BN * 6 * 64;            // [BN]

    // Pass 1: stream knn_x to its output slot (warms L2; NT stores on output).
    const size_t n4 = szKnnX / 4;
    lga_copy_kernel<<<2048, 256, 0, stream>>>((const f4*)knn_x, (f4*)out_knnx, n4);

    // Pass 2: one wave32 per point does the binning + f32-WMMA aggregation.
    lga_point_kernel<<<(unsigned)BN, 32, 0, stream>>>(
        lc_xyz, knn_xyz, knn_x, sphere, out_dp, out_ad, out_af, out_ki);
}